// Quant_ConvBn_skip_31061203484938
// MI455X (gfx1250) — compile-verified
//
#include <hip/hip_runtime.h>

// Problem constants (T,B,C,H,W) = (4,32,256,32,32), fp32 end to end.
#define T_    4
#define B_    32
#define C_    256
#define H_    32
#define W_    32
#define HW_   (H_ * W_)            // 1024
#define TILE_W 34
#define TILE_ELEMS (TILE_W * TILE_W)  // 1156
#define CNT_  ((float)(T_ * B_ * HW_))  // 131072 elements per channel per BN
#define EPS_  1e-5f

// ---------------------------------------------------------------------------
// gfx1250 async global->LDS path (ASYNCcnt-tracked). The builtin exists on
// this toolchain (round-1 diagnostic proved it); its pointer params are
// addrspace-qualified int*.
// ---------------------------------------------------------------------------
#if defined(__has_builtin)
#  if __has_builtin(__builtin_amdgcn_global_load_async_to_lds_b32)
#    define HAVE_ASYNC 1
#  endif
#endif
#ifndef HAVE_ASYNC
#  define HAVE_ASYNC 0
#endif

typedef __attribute__((address_space(1))) int g_int;
typedef __attribute__((address_space(3))) int l_int;

__device__ __forceinline__ void async_ld_b32(const float* g, float* l) {
#if HAVE_ASYNC
  __builtin_amdgcn_global_load_async_to_lds_b32(
      (g_int*)(void*)g,
      (l_int*)(void*)l,
      0, 0);
#else
  *l = *g;
#endif
}

__device__ __forceinline__ void wait_async() {
#if HAVE_ASYNC
#  if __has_builtin(__builtin_amdgcn_s_wait_asynccnt)
  __builtin_amdgcn_s_wait_asynccnt(0);
#  else
  asm volatile("s_wait_asynccnt 0" ::: "memory");
#  endif
#endif
}

// Load one 32x32 plane into a 34x34 LDS tile with zero halo.
// Halo cells are written with DS stores (disjoint from async targets);
// interior rows stream in via async-to-LDS. Caller must wait_async() +
// __syncthreads() before reading the tile.
__device__ __forceinline__ void load_tile(const float* __restrict__ xplane,
                                          float* tile) {
  const int tid = threadIdx.x;
  for (int i = tid; i < TILE_ELEMS; i += 256) {
    int r = i / TILE_W;
    int cc = i - r * TILE_W;
    if (r == 0 || r == (TILE_W - 1) || cc == 0 || cc == (TILE_W - 1))
      tile[i] = 0.0f;
  }
  const int pix = tid * 4;            // 4 consecutive pixels, same row
  const int row = pix >> 5;
  const int col = pix & 31;
  const float* gsrc = xplane + pix;
  float* ldst = tile + (row + 1) * TILE_W + (col + 1);
#pragma unroll
  for (int j = 0; j < 4; ++j) async_ld_b32(gsrc + j, ldst + j);
}

__device__ __forceinline__ float conv3x3(const float* t, int row, int col,
                                         const float* w9) {
  const float* r0 = t + row * TILE_W + col;
  const float* r1 = r0 + TILE_W;
  const float* r2 = r1 + TILE_W;
  return fmaf(w9[0], r0[0], fmaf(w9[1], r0[1], fmaf(w9[2], r0[2],
         fmaf(w9[3], r1[0], fmaf(w9[4], r1[1], fmaf(w9[5], r1[2],
         fmaf(w9[6], r2[0], fmaf(w9[7], r2[1], w9[8] * r2[2]))))))));
}

// Block reduce two accumulators at once (256 threads).
__device__ __forceinline__ void block_reduce2(float s, float q, float* red,
                                              float* outS, float* outQ) {
  const int tid = threadIdx.x;
  red[tid] = s;
  red[tid + 256] = q;
  __syncthreads();
  for (int st = 128; st > 0; st >>= 1) {
    if (tid < st) {
      red[tid] += red[tid + st];
      red[tid + 256] += red[tid + 256 + st];
    }
    __syncthreads();
  }
  *outS = red[0];
  *outQ = red[256];
}

// ---------------------------------------------------------------------------
// Pass 0: zero BN accumulators (graph replays re-run this).
// ---------------------------------------------------------------------------
__global__ void k_zero(float* __restrict__ p, int n) {
  int i = blockIdx.x * 256 + threadIdx.x;
  if (i < n) p[i] = 0.0f;
}

// ---------------------------------------------------------------------------
// Pass 1: depthwise conv + bias, accumulate per-channel sum / sumsq for BN1.
// One (t,b,c) plane per block.
// ---------------------------------------------------------------------------
__global__ __launch_bounds__(256)
void k_conv_stats(const float* __restrict__ x, const float* __restrict__ w,
                  const float* __restrict__ bias,
                  float* __restrict__ sum1, float* __restrict__ sq1) {
  __shared__ float tile[TILE_ELEMS];
  __shared__ float red[512];
  const int p = blockIdx.x;                 // (t*B + b)*C + c
  const int c = p & (C_ - 1);
  const float* xp = x + (size_t)p * HW_;
  __builtin_prefetch(xp, 0, 3);             // global_prefetch_b8

  load_tile(xp, tile);
  wait_async();
  __syncthreads();

  float w9[9];
#pragma unroll
  for (int k = 0; k < 9; ++k) w9[k] = w[c * 9 + k];
  const float bc = bias[c];

  const int tid = threadIdx.x;
  const int pix = tid * 4;
  const int row = pix >> 5;
  const int col = pix & 31;
  float s = 0.0f, q = 0.0f;
#pragma unroll
  for (int j = 0; j < 4; ++j) {
    float y = bc + conv3x3(tile, row, col + j, w9);
    s += y;
    q += y * y;
  }
  float bs, bq;
  block_reduce2(s, q, red, &bs, &bq);
  if (tid == 0) {
    atomicAdd(&sum1[c], bs);
    atomicAdd(&sq1[c], bq);
  }
}

// ---------------------------------------------------------------------------
// Pass 2/4: fold BN stats into per-channel scale/shift (one FMA at apply).
// ---------------------------------------------------------------------------
__global__ void k_finalize(const float* __restrict__ sum,
                           const float* __restrict__ sq,
                           const float* __restrict__ g,
                           const float* __restrict__ b,
                           float* __restrict__ scale,
                           float* __restrict__ shift) {
  const int c = threadIdx.x;                // 256 threads, 1 block
  const float inv = 1.0f / CNT_;
  const float m = sum[c] * inv;
  const float v = sq[c] * inv - m * m;      // biased variance (torch)
  const float r = rsqrtf(v + EPS_);
  const float sc = r * g[c];
  scale[c] = sc;
  shift[c] = b[c] - m * sc;
}

// ---------------------------------------------------------------------------
// Pass 3: recompute conv, apply BN1, run 4-step LIF in registers,
// z = spike + x, write z, accumulate BN2 stats. One (b,c) column per block.
// ---------------------------------------------------------------------------
__global__ __launch_bounds__(256)
void k_lif(const float* __restrict__ x, const float* __restrict__ w,
           const float* __restrict__ bias,
           const float* __restrict__ scale1, const float* __restrict__ shift1,
           float* __restrict__ out,
           float* __restrict__ sum2, float* __restrict__ sq2) {
  __shared__ float tile[T_][TILE_ELEMS];
  __shared__ float red[512];
  const int p = blockIdx.x;                 // b*C + c
  const int c = p & (C_ - 1);
  const int b = p >> 8;

  size_t plane[T_];
#pragma unroll
  for (int t = 0; t < T_; ++t) {
    plane[t] = ((size_t)(t * B_ + b) * C_ + c) * HW_;
    __builtin_prefetch(x + plane[t], 0, 3);
  }
#pragma unroll
  for (int t = 0; t < T_; ++t) load_tile(x + plane[t], tile[t]);
  wait_async();
  __syncthreads();

  float w9[9];
#pragma unroll
  for (int k = 0; k < 9; ++k) w9[k] = w[c * 9 + k];
  const float bc = bias[c];
  const float sc1 = scale1[c];
  const float sh1 = shift1[c];

  const int tid = threadIdx.x;
  const int pix0 = tid * 4;
  const int row = pix0 >> 5;
  const int col0 = pix0 & 31;
  float s = 0.0f, q = 0.0f;
#pragma unroll
  for (int j = 0; j < 4; ++j) {
    const int col = col0 + j;
    float v = 0.0f;                          // LIF membrane (v_reset = 0)
#pragma unroll
    for (int t = 0; t < T_; ++t) {
      float y = bc + conv3x3(tile[t], row, col, w9);
      y = fmaf(y, sc1, sh1);                 // BN1
      const float h = 0.5f * (v + y);        // v + (y - v)/tau, tau = 2
      const float sp = (h >= 1.0f) ? 1.0f : 0.0f;  // heaviside(h - vth)
      v = h * (1.0f - sp);                   // hard reset
      const float z = sp + tile[t][(row + 1) * TILE_W + (col + 1)];  // + x
      out[plane[t] + pix0 + j] = z;
      s += z;
      q += z * z;
    }
  }
  float bs, bq;
  block_reduce2(s, q, red, &bs, &bq);
  if (tid == 0) {
    atomicAdd(&sum2[c], bs);
    atomicAdd(&sq2[c], bq);
  }
}

// ---------------------------------------------------------------------------
// Pass 5: in-place BN2 apply, float4 per thread.
// ---------------------------------------------------------------------------
__global__ __launch_bounds__(256)
void k_bn2(float* __restrict__ out, const float* __restrict__ scale2,
           const float* __restrict__ shift2) {
  const size_t gid = (size_t)blockIdx.x * 256 + threadIdx.x;
  const size_t idx = gid * 4;
  const int c = (int)((idx >> 10) & (C_ - 1));  // plane = idx / HW; c = plane % C
  float4* p4 = (float4*)(out + idx);
  float4 v = *p4;
  const float sc = scale2[c];
  const float sh = shift2[c];
  v.x = fmaf(v.x, sc, sh);
  v.y = fmaf(v.y, sc, sh);
  v.z = fmaf(v.z, sc, sh);
  v.w = fmaf(v.w, sc, sh);
  *p4 = v;
}

// ---------------------------------------------------------------------------
extern "C" void kernel_launch(void* const* d_in, const int* in_sizes, int n_in,
                              void* d_out, int out_size, void* d_ws,
                              size_t ws_size, hipStream_t stream) {
  const float* x    = (const float*)d_in[0];
  const float* w    = (const float*)d_in[1];
  const float* bias = (const float*)d_in[2];
  const float* g1   = (const float*)d_in[3];
  const float* b1   = (const float*)d_in[4];
  const float* g2   = (const float*)d_in[5];
  const float* b2   = (const float*)d_in[6];
  float* out = (float*)d_out;

  float* ws = (float*)d_ws;
  float* sum1   = ws;
  float* sq1    = ws + 256;
  float* sum2   = ws + 512;
  float* sq2    = ws + 768;
  float* scale1 = ws + 1024;
  float* shift1 = ws + 1280;
  float* scale2 = ws + 1536;
  float* shift2 = ws + 1792;

  k_zero<<<4, 256, 0, stream>>>(ws, 1024);
  k_conv_stats<<<T_ * B_ * C_, 256, 0, stream>>>(x, w, bias, sum1, sq1);
  k_finalize<<<1, 256, 0, stream>>>(sum1, sq1, g1, b1, scale1, shift1);
  k_lif<<<B_ * C_, 256, 0, stream>>>(x, w, bias, scale1, shift1, out, sum2, sq2);
  k_finalize<<<1, 256, 0, stream>>>(sum2, sq2, g2, b2, scale2, shift2);
  k_bn2<<<(T_ * B_ * C_ * HW_) / 1024, 256, 0, stream>>>(out, scale2, shift2);
}